// AttentionalPlanarRemappingOld_75634374082661
// MI455X (gfx1250) — compile-verified
//
#include <hip/hip_runtime.h>
#include <math.h>

typedef __attribute__((ext_vector_type(2))) float v2f;
typedef __attribute__((ext_vector_type(8))) float v8f;

#define N_BATCH 32
#define C_DIM   512   // rows of W, also K dimension
#define HW      1024  // 32*32 flattened spatial = N dimension of GEMM

// ---------------------------------------------------------------------------
// Pass 1: per-row softmax stats. One wave32 per row of atts (length 512).
// stats[2*row] = rowmax, stats[2*row+1] = 1/sum(exp(x-max)).
// ---------------------------------------------------------------------------
__global__ __launch_bounds__(256) void apr_softmax_stats(
    const float* __restrict__ atts, float* __restrict__ stats) {
  const int row  = blockIdx.x * 8 + (threadIdx.x >> 5);
  const int lane = threadIdx.x & 31;
  const float4* p = reinterpret_cast<const float4*>(atts + (size_t)row * C_DIM);

  float4 v[4];
  float mx = -3.402823466e+38f;
#pragma unroll
  for (int i = 0; i < 4; ++i) {
    v[i] = p[i * 32 + lane];  // coalesced: 512B per access
    mx = fmaxf(mx, fmaxf(fmaxf(v[i].x, v[i].y), fmaxf(v[i].z, v[i].w)));
  }
#pragma unroll
  for (int off = 16; off > 0; off >>= 1)
    mx = fmaxf(mx, __shfl_xor(mx, off, 32));

  float s = 0.f;
#pragma unroll
  for (int i = 0; i < 4; ++i) {
    s += __expf(v[i].x - mx) + __expf(v[i].y - mx) +
         __expf(v[i].z - mx) + __expf(v[i].w - mx);
  }
#pragma unroll
  for (int off = 16; off > 0; off >>= 1)
    s += __shfl_xor(s, off, 32);

  if (lane == 0) {
    stats[2 * row]     = mx;
    stats[2 * row + 1] = 1.0f / s;
  }
}

// ---------------------------------------------------------------------------
// Pass 2: out[n] = softmax(atts[n]) x images[n]
//   Block tile: BM=64 x BN=256, BK=32, double-buffered LDS pipeline.
//   256 threads = 8 waves (2M x 4N); each wave owns a 32x64 strip =
//   8 accumulators, 8 V_WMMA_F32_16X16X4_F32 per K-step.
//   Next chunk's global loads issue BEFORE the WMMA stream (latency hidden).
// ---------------------------------------------------------------------------
__global__ __launch_bounds__(256) void apr_gemm(
    const float* __restrict__ atts, const float* __restrict__ images,
    const float* __restrict__ stats, float* __restrict__ out) {
  constexpr int BM = 64, BN = 256, BK = 32, NCHUNK = C_DIM / BK;
  // A: row stride 34 floats -> 8B-aligned float2 reads, conflict-free
  __shared__ float As[2][BM][BK + 2];
  // B: Bs2[k/2][col] = {B[k][col], B[k+1][col]}; row stride 272 float2
  //    (= 544 words === 32 mod 64 banks -> lo/hi half-wave banks disjoint)
  __shared__ v2f Bs2[2][BK / 2][BN + 16];

  const int n      = blockIdx.z;
  const int cBlock = blockIdx.y * BM;
  const int nBlock = blockIdx.x * BN;
  const int tid    = threadIdx.x;
  const int lane   = tid & 31;
  const int wave   = tid >> 5;
  const int waveM  = wave & 1;    // 2 strips of 32 rows
  const int waveN  = wave >> 1;   // 4 strips of 64 cols
  const int hi     = lane >> 4;   // half-wave selector
  const int l15    = lane & 15;

  // --- A-tile loader mapping (fused softmax on stage) ---
  const int a_kq = tid & 7;        // float4 along K (k = a_kq*4)
  const int a_r  = tid >> 3;       // row 0..31, plus row+32
  const int rowg0 = n * C_DIM + cBlock + a_r;
  const int rowg1 = rowg0 + 32;
  const float mx0 = stats[2 * rowg0], inv0 = stats[2 * rowg0 + 1];
  const float mx1 = stats[2 * rowg1], inv1 = stats[2 * rowg1 + 1];
  const float* aP0 = atts + (size_t)rowg0 * C_DIM + a_kq * 4;
  const float* aP1 = atts + (size_t)rowg1 * C_DIM + a_kq * 4;

  // --- B-tile loader mapping: rows (2*rp, 2*rp+1) x float4 of cols ---
  const int b_c4 = tid & 63;       // float4 col group (256/4 = 64)
  const int b_rp = tid >> 6;       // K row-pair 0..3, step 4 (BK/2 = 16 pairs)
  const float* bP = images + (size_t)(n * C_DIM) * HW + nBlock + b_c4 * 4;

  // staged registers for the in-flight chunk
  float4 sx0, sx1, sy0[4], sy1[4];

  auto loadGlobal = [&](int kb) {
    sx0 = *reinterpret_cast<const float4*>(aP0 + kb);
    sx1 = *reinterpret_cast<const float4*>(aP1 + kb);
#pragma unroll
    for (int i = 0; i < 4; ++i) {
      const int kr = kb + 2 * (b_rp + i * 4);
      sy0[i] = *reinterpret_cast<const float4*>(bP + (size_t)(kr + 0) * HW);
      sy1[i] = *reinterpret_cast<const float4*>(bP + (size_t)(kr + 1) * HW);
    }
  };

  auto stageLDS = [&](int buf) {
    As[buf][a_r][a_kq * 4 + 0] = __expf(sx0.x - mx0) * inv0;
    As[buf][a_r][a_kq * 4 + 1] = __expf(sx0.y - mx0) * inv0;
    As[buf][a_r][a_kq * 4 + 2] = __expf(sx0.z - mx0) * inv0;
    As[buf][a_r][a_kq * 4 + 3] = __expf(sx0.w - mx0) * inv0;
    As[buf][a_r + 32][a_kq * 4 + 0] = __expf(sx1.x - mx1) * inv1;
    As[buf][a_r + 32][a_kq * 4 + 1] = __expf(sx1.y - mx1) * inv1;
    As[buf][a_r + 32][a_kq * 4 + 2] = __expf(sx1.z - mx1) * inv1;
    As[buf][a_r + 32][a_kq * 4 + 3] = __expf(sx1.w - mx1) * inv1;
#pragma unroll
    for (int i = 0; i < 4; ++i) {
      const int rp = b_rp + i * 4;  // pair index 0..15
      Bs2[buf][rp][b_c4 * 4 + 0] = v2f{sy0[i].x, sy1[i].x};
      Bs2[buf][rp][b_c4 * 4 + 1] = v2f{sy0[i].y, sy1[i].y};
      Bs2[buf][rp][b_c4 * 4 + 2] = v2f{sy0[i].z, sy1[i].z};
      Bs2[buf][rp][b_c4 * 4 + 3] = v2f{sy0[i].w, sy1[i].w};
    }
  };

  v8f acc[2][4] = {};

  // prologue: stage chunk 0
  loadGlobal(0);
  stageLDS(0);
  __syncthreads();

  for (int kc = 0; kc < NCHUNK; ++kc) {
    const int cur = kc & 1;
    const bool more = (kc + 1) < NCHUNK;
    if (more) loadGlobal((kc + 1) * BK);  // issue global loads early

    // Math: 8 K-steps x (2 M-subtiles x 4 N-subtiles) WMMAs on buffer `cur`
#pragma unroll
    for (int k = 0; k < BK; k += 4) {
      // A frag 16x4: lane<16 -> (K,K+1), lane>=16 -> (K+2,K+3), row M = l15
      v2f a0, a1;
      const float* ar0 = &As[cur][waveM * 32 + l15][k + 2 * hi];
      const float* ar1 = &As[cur][waveM * 32 + 16 + l15][k + 2 * hi];
      a0[0] = ar0[0];
      a0[1] = ar0[1];
      a1[0] = ar1[0];
      a1[1] = ar1[1];
#pragma unroll
      for (int t = 0; t < 4; ++t) {
        // B frag 4x16 in one b64: row pair k/2 (lo lanes) / k/2+1 (hi lanes)
        const v2f b = Bs2[cur][(k >> 1) + hi][waveN * 64 + t * 16 + l15];
        acc[0][t] = __builtin_amdgcn_wmma_f32_16x16x4_f32(
            false, a0, false, b, (short)0, acc[0][t], false, false);
        acc[1][t] = __builtin_amdgcn_wmma_f32_16x16x4_f32(
            false, a1, false, b, (short)0, acc[1][t], false, false);
      }
    }

    if (more) stageLDS(cur ^ 1);  // fill the other buffer behind the math
    __syncthreads();
  }

  // Write back: C/D layout -> VGPR v holds row M = v + 8*hi, col N = l15
#pragma unroll
  for (int s = 0; s < 2; ++s) {
    const int orow = cBlock + waveM * 32 + s * 16 + hi * 8;
#pragma unroll
    for (int t = 0; t < 4; ++t) {
      const int ocol = nBlock + waveN * 64 + t * 16 + l15;
      float* op = out + ((size_t)(n * C_DIM + orow)) * HW + ocol;
#pragma unroll
      for (int v = 0; v < 8; ++v) {
        op[(size_t)v * HW] = acc[s][t][v];
      }
    }
  }
}

// ---------------------------------------------------------------------------
extern "C" void kernel_launch(void* const* d_in, const int* in_sizes, int n_in,
                              void* d_out, int out_size, void* d_ws, size_t ws_size,
                              hipStream_t stream) {
  const float* images = (const float*)d_in[0];  // [32,512,32,32]
  const float* atts   = (const float*)d_in[1];  // [32,512,512]
  float* out   = (float*)d_out;                 // [32,512,32,32]
  float* stats = (float*)d_ws;                  // [32*512][2] = 128 KB

  // Pass 1: 16384 rows, one wave per row, 8 rows per 256-thread block.
  apr_softmax_stats<<<(N_BATCH * C_DIM) / 8, 256, 0, stream>>>(atts, stats);

  // Pass 2: grid = (N tiles, M tiles, batch) = (1024/256, 512/64, 32).
  dim3 grid(HW / 256, C_DIM / 64, N_BATCH);
  apr_gemm<<<grid, 256, 0, stream>>>(atts, images, stats, out);
}